// GraphSAGEBipartiteWithAttention_35648228556941
// MI455X (gfx1250) — compile-verified
//
#include <hip/hip_runtime.h>
#include <math.h>

// Problem constants (match reference)
#define NGg 20000
#define NSs 2000
#define FXx 256
#define FYy 256
#define NEe 1000000
#define EDd 96
#define KDd 48

#define SLICES_G 2
#define SLICES_S 16

typedef float v2f __attribute__((ext_vector_type(2)));
typedef float v8f __attribute__((ext_vector_type(8)));

// ---------------------------------------------------------------------------
// fp32 WMMA: D[16x16] = A[16x4] * B[4x16] + C   (V_WMMA_F32_16X16X4_F32)
// ---------------------------------------------------------------------------
__device__ __forceinline__ v8f wmma_f32x4(v2f a, v2f b, v8f c) {
  return __builtin_amdgcn_wmma_f32_16x16x4_f32(
      /*neg_a=*/false, a, /*neg_b=*/false, b,
      /*c_mod=*/(short)0, c, /*reuse_a=*/false, /*reuse_b=*/false);
}

__device__ __forceinline__ void atomicAddF(float* addr, float v) {
  unsafeAtomicAdd(addr, v);  // native global_atomic_add_f32
}

// ---------------------------------------------------------------------------
// SpMM scatter: out[rows[e], :] += vals[e] * X[cols[e], :]   (out pre-zeroed)
// ---------------------------------------------------------------------------
__global__ __launch_bounds__(256) void spmm_scatter(
    const float* __restrict__ vals, const int* __restrict__ rows,
    const int* __restrict__ cols, const float* __restrict__ X,
    float* __restrict__ out, int E, int F) {
  int e = blockIdx.x * (blockDim.x >> 5) + (threadIdx.x >> 5);
  if (e >= E) return;
  int lane = threadIdx.x & 31;
  int r = rows[e];
  int c = cols[e];
  float v = vals[e];
  const float* xs = X + (size_t)c * F;
  float* os = out + (size_t)r * F;
  for (int f = lane; f < F; f += 32) atomicAddF(os + f, v * xs[f]);
}

// ---------------------------------------------------------------------------
// Row-strip dual GEMM (compile-time specialized): out = act(A*WA [+ B*WB] + bias)
// N = NT*16. K consumed 8 at a time: all fragments for two k-steps are loaded
// into registers first, then 2*NT straight-line WMMAs, so loads clause up and
// s_wait_loadcnt staggers across the WMMA stream instead of serializing.
// ---------------------------------------------------------------------------
template <int NT, bool DUAL, bool RELU, bool BIAS>
__global__ __launch_bounds__(128) void gemm_rowstrip(
    const float* __restrict__ A, const float* __restrict__ WA,
    const float* __restrict__ B, const float* __restrict__ WB,
    const float* __restrict__ bias, float* __restrict__ out, int M, int K) {
  const int N = NT * 16;
  const int wid = threadIdx.x >> 5;
  const int m0 = (blockIdx.x * 4 + wid) * 16;
  if (m0 >= M) return;
  const int lane = threadIdx.x & 31;
  const int r = lane & 15;
  const int h = lane >> 4;
  v8f acc[NT] = {};
  for (int k0 = 0; k0 < K; k0 += 8) {
    // ---- load phase: 2 A fragments + 2*NT B fragments ----
    v2f a0, a1, b0[NT], b1[NT];
    const float* ar = A + (size_t)(m0 + r) * K + k0 + 2 * h;
    a0.x = ar[0];
    a0.y = ar[1];
    a1.x = ar[4];
    a1.y = ar[5];
    const float* w00 = WA + (size_t)(k0 + 2 * h) * N + r;
    const float* w01 = WA + (size_t)(k0 + 2 * h + 1) * N + r;
    const float* w10 = WA + (size_t)(k0 + 4 + 2 * h) * N + r;
    const float* w11 = WA + (size_t)(k0 + 4 + 2 * h + 1) * N + r;
#pragma unroll
    for (int t = 0; t < NT; ++t) {
      b0[t].x = w00[t * 16];
      b0[t].y = w01[t * 16];
      b1[t].x = w10[t * 16];
      b1[t].y = w11[t * 16];
    }
    // ---- compute phase ----
#pragma unroll
    for (int t = 0; t < NT; ++t) acc[t] = wmma_f32x4(a0, b0[t], acc[t]);
#pragma unroll
    for (int t = 0; t < NT; ++t) acc[t] = wmma_f32x4(a1, b1[t], acc[t]);
  }
  if (DUAL) {
    for (int k0 = 0; k0 < K; k0 += 8) {
      v2f a0, a1, b0[NT], b1[NT];
      const float* ar = B + (size_t)(m0 + r) * K + k0 + 2 * h;
      a0.x = ar[0];
      a0.y = ar[1];
      a1.x = ar[4];
      a1.y = ar[5];
      const float* w00 = WB + (size_t)(k0 + 2 * h) * N + r;
      const float* w01 = WB + (size_t)(k0 + 2 * h + 1) * N + r;
      const float* w10 = WB + (size_t)(k0 + 4 + 2 * h) * N + r;
      const float* w11 = WB + (size_t)(k0 + 4 + 2 * h + 1) * N + r;
#pragma unroll
      for (int t = 0; t < NT; ++t) {
        b0[t].x = w00[t * 16];
        b0[t].y = w01[t * 16];
        b1[t].x = w10[t * 16];
        b1[t].y = w11[t * 16];
      }
#pragma unroll
      for (int t = 0; t < NT; ++t) acc[t] = wmma_f32x4(a0, b0[t], acc[t]);
#pragma unroll
      for (int t = 0; t < NT; ++t) acc[t] = wmma_f32x4(a1, b1[t], acc[t]);
    }
  }
#pragma unroll
  for (int t = 0; t < NT; ++t) {
    float bv = BIAS ? bias[t * 16 + r] : 0.0f;
#pragma unroll
    for (int i = 0; i < 8; ++i) {
      float v = acc[t][i] + bv;
      if (RELU) v = fmaxf(v, 0.0f);
      out[(size_t)(m0 + i + 8 * h) * N + t * 16 + r] = v;
    }
  }
}

template <int NT, bool DUAL, bool RELU, bool BIAS>
static void launch_gemm(const float* A, const float* WA, const float* B,
                        const float* WB, const float* bias, float* out, int M,
                        int K, hipStream_t stream) {
  int blocks = (M / 16 + 3) / 4;
  gemm_rowstrip<NT, DUAL, RELU, BIAS>
      <<<blocks, 128, 0, stream>>>(A, WA, B, WB, bias, out, M, K);
}

// ---------------------------------------------------------------------------
// Split-K flash attention, phase 1.
// Block (mtile, slice): streams key tiles kt = slice, slice+nsl, ...
// All K / V fragments for a tile are preloaded into register arrays before
// the WMMA chains, overlapping the 24 loads with the accumulate stream.
// ---------------------------------------------------------------------------
__global__ __launch_bounds__(32) void flash_attn_part(
    const float* __restrict__ Q, const float* __restrict__ Km,
    const float* __restrict__ V, float* __restrict__ Opart,
    float* __restrict__ mpart, float* __restrict__ lpart, int M, int Nk,
    int nsl) {
  __shared__ float Plds[16][17];
  const int m0 = blockIdx.x * 16;
  const int slice = blockIdx.y;
  const int lane = threadIdx.x & 31;
  const int r = lane & 15;
  const int h = lane >> 4;
  const float scale = 0.102062072615966f;  // 1/sqrt(96)
  const int nkt = Nk >> 4;

  // Q fragments for the whole 16x96 tile (24 x v2f)
  v2f qf[EDd / 4];
#pragma unroll
  for (int k0 = 0; k0 < EDd; k0 += 4) {
    const float* qr = Q + (size_t)(m0 + r) * EDd + k0 + 2 * h;
    qf[k0 / 4].x = qr[0];
    qf[k0 / 4].y = qr[1];
  }

  v8f O[6] = {};
  float mrow[8], lrow[8];
#pragma unroll
  for (int i = 0; i < 8; ++i) {
    mrow[i] = -1e30f;
    lrow[i] = 0.0f;
  }

  for (int kt = slice; kt < nkt; kt += nsl) {
    const int kn0 = kt * 16;
    // Prefetch the next tile of K and V for this slice (global_prefetch_b8)
    const int ktn = kt + nsl;
    if (ktn < nkt) {
      __builtin_prefetch(Km + (size_t)(ktn * 16 + r) * EDd + h * 48, 0, 3);
      __builtin_prefetch(V + (size_t)(ktn * 16 + r) * EDd + h * 48, 0, 3);
    }
    // ---- load phase: all 24 K fragments (B(k, j) = K[j, k]) ----
    v2f kb[EDd / 4];
#pragma unroll
    for (int k0 = 0; k0 < EDd; k0 += 4) {
      const float* kr = Km + (size_t)(kn0 + r) * EDd + k0 + 2 * h;
      kb[k0 / 4].x = kr[0];
      kb[k0 / 4].y = kr[1];
    }
    // ---- compute phase: S = Q * K^T ----
    v8f S = {};
#pragma unroll
    for (int i = 0; i < EDd / 4; ++i) S = wmma_f32x4(qf[i], kb[i], S);
    // Online softmax: row reductions across the 16 lanes of each half-group
#pragma unroll
    for (int i = 0; i < 8; ++i) {
      float s = S[i] * scale;
      float rm = s;
#pragma unroll
      for (int mk = 8; mk >= 1; mk >>= 1)
        rm = fmaxf(rm, __shfl_xor(rm, mk, 32));
      float mnew = fmaxf(mrow[i], rm);
      float alpha = __expf(mrow[i] - mnew);
      float pe = __expf(s - mnew);
      float rs = pe;
#pragma unroll
      for (int mk = 8; mk >= 1; mk >>= 1) rs += __shfl_xor(rs, mk, 32);
      lrow[i] = lrow[i] * alpha + rs;
      mrow[i] = mnew;
#pragma unroll
      for (int g = 0; g < 6; ++g) O[g][i] *= alpha;
      Plds[i + 8 * h][r] = pe;  // C-layout -> LDS
    }
    __syncthreads();
    // Reload P in A-layout (16x16, K = 16 keys)
    v2f pa[4];
#pragma unroll
    for (int kk = 0; kk < 16; kk += 4) {
      pa[kk / 4].x = Plds[r][kk + 2 * h];
      pa[kk / 4].y = Plds[r][kk + 2 * h + 1];
    }
    // ---- load phase: all 24 V fragments ----
    v2f vbuf[4][6];
#pragma unroll
    for (int kk = 0; kk < 16; kk += 4) {
      const float* v0 = V + (size_t)(kn0 + kk + 2 * h) * EDd + r;
      const float* v1 = V + (size_t)(kn0 + kk + 2 * h + 1) * EDd + r;
#pragma unroll
      for (int g = 0; g < 6; ++g) {
        vbuf[kk / 4][g].x = v0[g * 16];
        vbuf[kk / 4][g].y = v1[g * 16];
      }
    }
    // ---- compute phase: O += P * V ----
#pragma unroll
    for (int g = 0; g < 6; ++g) {
#pragma unroll
      for (int kk = 0; kk < 4; ++kk)
        O[g] = wmma_f32x4(pa[kk], vbuf[kk][g], O[g]);
    }
    __syncthreads();
  }
  // store partial (unnormalized) O and per-row m, l
#pragma unroll
  for (int i = 0; i < 8; ++i) {
    const size_t row = (size_t)slice * M + m0 + i + 8 * h;
#pragma unroll
    for (int g = 0; g < 6; ++g)
      Opart[row * EDd + g * 16 + r] = O[g][i];
    if (r == 0) {
      mpart[row] = mrow[i];
      lpart[row] = lrow[i];
    }
  }
}

// ---------------------------------------------------------------------------
// Split-K flash attention, phase 2: merge slices with log-sum-exp rescale.
// ---------------------------------------------------------------------------
__global__ __launch_bounds__(256) void flash_attn_merge(
    const float* __restrict__ Opart, const float* __restrict__ mpart,
    const float* __restrict__ lpart, float* __restrict__ out, int M,
    int nsl) {
  int idx = blockIdx.x * blockDim.x + threadIdx.x;
  if (idx >= M * EDd) return;
  int row = idx / EDd;
  int d = idx - row * EDd;
  float mg = -1e30f;
  for (int s = 0; s < nsl; ++s) mg = fmaxf(mg, mpart[(size_t)s * M + row]);
  float l = 0.0f, o = 0.0f;
  for (int s = 0; s < nsl; ++s) {
    float wgt = __expf(mpart[(size_t)s * M + row] - mg);
    l += lpart[(size_t)s * M + row] * wgt;
    o += Opart[((size_t)s * M + row) * EDd + d] * wgt;
  }
  out[(size_t)row * EDd + d] = o / l;
}

// ---------------------------------------------------------------------------
// Row L2-normalize (F = 48): one wave per row
// ---------------------------------------------------------------------------
__global__ __launch_bounds__(256) void rownorm(const float* __restrict__ in,
                                               float* __restrict__ out, int M,
                                               int F) {
  int row = blockIdx.x * (blockDim.x >> 5) + (threadIdx.x >> 5);
  if (row >= M) return;
  int lane = threadIdx.x & 31;
  float v0 = (lane < F) ? in[(size_t)row * F + lane] : 0.0f;
  float v1 = (lane + 32 < F) ? in[(size_t)row * F + lane + 32] : 0.0f;
  float s = v0 * v0 + v1 * v1;
#pragma unroll
  for (int mk = 16; mk >= 1; mk >>= 1) s += __shfl_xor(s, mk, 32);
  float inv = 1.0f / (sqrtf(s) + 1e-6f);
  if (lane < F) out[(size_t)row * F + lane] = v0 * inv;
  if (lane + 32 < F) out[(size_t)row * F + lane + 32] = v1 * inv;
}

// ---------------------------------------------------------------------------
// Cosine decoder: out[m,n] = 0.5*(dot(xn[m], yn[n]) + 1), K = 48
// ---------------------------------------------------------------------------
__global__ __launch_bounds__(32) void cos_decoder(
    const float* __restrict__ xn, const float* __restrict__ yn,
    float* __restrict__ out, int M, int N) {
  const int n0 = blockIdx.x * 16;
  const int m0 = blockIdx.y * 16;
  const int lane = threadIdx.x & 31;
  const int r = lane & 15;
  const int h = lane >> 4;
  // ---- load phase: all 12 A and 12 B fragments ----
  v2f af[KDd / 4], bf[KDd / 4];
#pragma unroll
  for (int k0 = 0; k0 < KDd; k0 += 4) {
    const float* ar = xn + (size_t)(m0 + r) * KDd + k0 + 2 * h;
    af[k0 / 4].x = ar[0];
    af[k0 / 4].y = ar[1];
    const float* br = yn + (size_t)(n0 + r) * KDd + k0 + 2 * h;  // B = yn^T
    bf[k0 / 4].x = br[0];
    bf[k0 / 4].y = br[1];
  }
  // ---- compute phase ----
  v8f acc = {};
#pragma unroll
  for (int i = 0; i < KDd / 4; ++i) acc = wmma_f32x4(af[i], bf[i], acc);
#pragma unroll
  for (int i = 0; i < 8; ++i)
    out[(size_t)(m0 + i + 8 * h) * N + n0 + r] = (acc[i] + 1.0f) * 0.5f;
}

// ---------------------------------------------------------------------------
// Host orchestration
// ---------------------------------------------------------------------------
extern "C" void kernel_launch(void* const* d_in, const int* in_sizes, int n_in,
                              void* d_out, int out_size, void* d_ws,
                              size_t ws_size, hipStream_t stream) {
  const float* x = (const float*)d_in[0];
  const float* y = (const float*)d_in[1];
  const int* edge_row = (const int*)d_in[2];
  const int* edge_col = (const int*)d_in[3];
  const float* vals_gs = (const float*)d_in[4];
  const float* vals_sg = (const float*)d_in[5];
  const float* Wg1_self = (const float*)d_in[6];
  const float* Wg1_neigh = (const float*)d_in[7];
  const float* Ws1_self = (const float*)d_in[8];
  const float* Ws1_neigh = (const float*)d_in[9];
  const float* Wg2_self = (const float*)d_in[10];
  const float* Wg2_neigh = (const float*)d_in[11];
  const float* Ws2_self = (const float*)d_in[12];
  const float* Ws2_neigh = (const float*)d_in[13];
  const float* WQ = (const float*)d_in[14];
  const float* bQ = (const float*)d_in[15];
  const float* WK = (const float*)d_in[16];
  const float* bK = (const float*)d_in[17];
  const float* WV = (const float*)d_in[18];
  const float* bV = (const float*)d_in[19];
  const float* Wpg = (const float*)d_in[20];
  const float* bpg = (const float*)d_in[21];
  const float* Wps = (const float*)d_in[22];
  const float* bps = (const float*)d_in[23];
  const float* Wlx = (const float*)d_in[24];
  const float* Wly = (const float*)d_in[25];
  float* out = (float*)d_out;

  // Workspace carve (floats)
  float* w = (float*)d_ws;
  size_t off = 0;
  auto alloc = [&](size_t n) {
    float* p = w + off;
    off += n;
    return p;
  };
  float* neigh_g = alloc((size_t)NGg * FYy);
  float* neigh_s = alloc((size_t)NSs * FXx);
  float* g1 = alloc((size_t)NGg * EDd);
  float* s1 = alloc((size_t)NSs * EDd);
  float* ng2 = alloc((size_t)NGg * EDd);
  float* ns2 = alloc((size_t)NSs * EDd);
  float* g2 = alloc((size_t)NGg * EDd);
  float* s2 = alloc((size_t)NSs * EDd);
  float* Qg = alloc((size_t)NGg * EDd);
  float* Kg = alloc((size_t)NGg * EDd);
  float* Vg = alloc((size_t)NGg * EDd);
  float* Qs = alloc((size_t)NSs * EDd);
  float* Ks = alloc((size_t)NSs * EDd);
  float* Vs = alloc((size_t)NSs * EDd);
  float* g2a = alloc((size_t)NGg * EDd);
  float* s2a = alloc((size_t)NSs * EDd);
  float* gp = alloc((size_t)NGg * EDd);
  float* sp = alloc((size_t)NSs * EDd);
  float* xk = alloc((size_t)NGg * KDd);
  float* yk = alloc((size_t)NSs * KDd);
  float* xn = alloc((size_t)NGg * KDd);
  float* yn = alloc((size_t)NSs * KDd);
  // split-K attention scratch
  float* Opart_g = alloc((size_t)SLICES_G * NGg * EDd);
  float* mpart_g = alloc((size_t)SLICES_G * NGg);
  float* lpart_g = alloc((size_t)SLICES_G * NGg);
  float* Opart_s = alloc((size_t)SLICES_S * NSs * EDd);
  float* mpart_s = alloc((size_t)SLICES_S * NSs);
  float* lpart_s = alloc((size_t)SLICES_S * NSs);
  (void)ws_size;
  (void)n_in;
  (void)in_sizes;
  (void)out_size;

  const int spmmBlocks = (NEe + 7) / 8;

  // ----- Layer 1 -----
  hipMemsetAsync(neigh_g, 0, (size_t)NGg * FYy * sizeof(float), stream);
  hipMemsetAsync(neigh_s, 0, (size_t)NSs * FXx * sizeof(float), stream);
  spmm_scatter<<<spmmBlocks, 256, 0, stream>>>(vals_gs, edge_row, edge_col, y,
                                               neigh_g, NEe, FYy);
  spmm_scatter<<<spmmBlocks, 256, 0, stream>>>(vals_sg, edge_col, edge_row, x,
                                               neigh_s, NEe, FXx);
  launch_gemm<6, true, true, false>(x, Wg1_self, neigh_g, Wg1_neigh, nullptr,
                                    g1, NGg, FXx, stream);
  launch_gemm<6, true, true, false>(y, Ws1_self, neigh_s, Ws1_neigh, nullptr,
                                    s1, NSs, FYy, stream);

  // ----- Layer 2 -----
  hipMemsetAsync(ng2, 0, (size_t)NGg * EDd * sizeof(float), stream);
  hipMemsetAsync(ns2, 0, (size_t)NSs * EDd * sizeof(float), stream);
  spmm_scatter<<<spmmBlocks, 256, 0, stream>>>(vals_gs, edge_row, edge_col, s1,
                                               ng2, NEe, EDd);
  spmm_scatter<<<spmmBlocks, 256, 0, stream>>>(vals_sg, edge_col, edge_row, g1,
                                               ns2, NEe, EDd);
  launch_gemm<6, true, true, false>(g1, Wg2_self, ng2, Wg2_neigh, nullptr, g2,
                                    NGg, EDd, stream);
  launch_gemm<6, true, true, false>(s1, Ws2_self, ns2, Ws2_neigh, nullptr, s2,
                                    NSs, EDd, stream);

  // ----- QKV (shared weights, both directions) -----
  launch_gemm<6, false, false, true>(g2, WQ, nullptr, nullptr, bQ, Qg, NGg,
                                     EDd, stream);
  launch_gemm<6, false, false, true>(g2, WK, nullptr, nullptr, bK, Kg, NGg,
                                     EDd, stream);
  launch_gemm<6, false, false, true>(g2, WV, nullptr, nullptr, bV, Vg, NGg,
                                     EDd, stream);
  launch_gemm<6, false, false, true>(s2, WQ, nullptr, nullptr, bQ, Qs, NSs,
                                     EDd, stream);
  launch_gemm<6, false, false, true>(s2, WK, nullptr, nullptr, bK, Ks, NSs,
                                     EDd, stream);
  launch_gemm<6, false, false, true>(s2, WV, nullptr, nullptr, bV, Vs, NSs,
                                     EDd, stream);

  // ----- Attention both ways (split-K flash) -----
  flash_attn_part<<<dim3(NGg / 16, SLICES_G), 32, 0, stream>>>(
      Qg, Ks, Vs, Opart_g, mpart_g, lpart_g, NGg, NSs, SLICES_G);
  flash_attn_merge<<<(NGg * EDd + 255) / 256, 256, 0, stream>>>(
      Opart_g, mpart_g, lpart_g, g2a, NGg, SLICES_G);
  flash_attn_part<<<dim3(NSs / 16, SLICES_S), 32, 0, stream>>>(
      Qs, Kg, Vg, Opart_s, mpart_s, lpart_s, NSs, NGg, SLICES_S);
  flash_attn_merge<<<(NSs * EDd + 255) / 256, 256, 0, stream>>>(
      Opart_s, mpart_s, lpart_s, s2a, NSs, SLICES_S);

  // ----- Projections -----
  launch_gemm<6, false, true, true>(g2a, Wpg, nullptr, nullptr, bpg, gp, NGg,
                                    EDd, stream);
  launch_gemm<6, false, true, true>(s2a, Wps, nullptr, nullptr, bps, sp, NSs,
                                    EDd, stream);

  // ----- Cosine decoder -----
  launch_gemm<3, false, false, false>(gp, Wlx, nullptr, nullptr, nullptr, xk,
                                      NGg, EDd, stream);
  launch_gemm<3, false, false, false>(sp, Wly, nullptr, nullptr, nullptr, yk,
                                      NSs, EDd, stream);
  rownorm<<<(NGg + 7) / 8, 256, 0, stream>>>(xk, xn, NGg, KDd);
  rownorm<<<(NSs + 7) / 8, 256, 0, stream>>>(yk, yn, NSs, KDd);
  cos_decoder<<<dim3(NSs / 16, NGg / 16), 32, 0, stream>>>(xn, yn, out, NGg,
                                                           NSs);
}